// WeightedBCEWithLogitsLoss_23373212025002
// MI455X (gfx1250) — compile-verified
//
#include <hip/hip_runtime.h>
#include <math.h>

#define Bn   8
#define Cn   6
#define Hn   512
#define Wn   512
#define HWn  (Hn * Wn)           // 262144
#define CHWn (Cn * HWn)          // 1572864
#define BHW  (Bn * HWn)          // 2097152
#define BCHW (Bn * Cn * HWn)     // 12582912
#define NBLK3 2048
#define ITER3 24                 // NBLK3*256*ITER3 == BCHW exactly (uniform EXEC)
#define EPSF 1e-10f

typedef __attribute__((ext_vector_type(2))) float v2f;
typedef __attribute__((ext_vector_type(8))) float v8f;

// ---------------- init global min/max (uint-punned, weights are >= 0) ------
__global__ void k_init(unsigned* minmax) {
    if (threadIdx.x == 0) {
        minmax[0] = 0x7F800000u;  // +inf bits  (running min)
        minmax[1] = 0x00000000u;  // +0.0 bits  (running max)
    }
}

// ---------------- channel sum: fw[b,h,w] = sum_c t ------------------------
__global__ void __launch_bounds__(256) k_channel(const int* __restrict__ tgt,
                                                 float* __restrict__ fw) {
    int i = blockIdx.x * 256 + threadIdx.x;      // pixel index in B*H*W
    if (i >= BHW) return;
    int b  = i / HWn;
    int hw = i - b * HWn;
    const int* p = tgt + (size_t)b * CHWn + hw;
    int s = 0;
#pragma unroll
    for (int c = 0; c < Cn; ++c) s += p[(size_t)c * HWn];
    fw[i] = (float)s;
}

// ---------------- 9x9 nearest-differing-pixel stencil ---------------------
__global__ void __launch_bounds__(256) k_stencil(const float* __restrict__ fw,
                                                 float* __restrict__ wraw,
                                                 unsigned* __restrict__ minmax) {
    __shared__ float tile[40][40];               // 32x32 tile + 4 halo
    __shared__ unsigned smin, smax;
    const int bx = blockIdx.x * 32, by = blockIdx.y * 32, bz = blockIdx.z;
    const float* fwb = fw + (size_t)bz * HWn;
    const int tid = threadIdx.y * 32 + threadIdx.x;
    if (tid == 0) { smin = 0x7F800000u; smax = 0u; }

    // load 40x40 binary tile with edge clamp (binarize fw on the fly)
    for (int t = tid; t < 1600; t += 256) {
        int ly = t / 40, lx = t - ly * 40;
        int gy = by + ly - 4; gy = gy < 0 ? 0 : (gy > Hn - 1 ? Hn - 1 : gy);
        int gx = bx + lx - 4; gx = gx < 0 ? 0 : (gx > Wn - 1 ? Wn - 1 : gx);
        tile[ly][lx] = (fwb[gy * Wn + gx] > 0.0f) ? 1.0f : 0.0f;
    }
    __syncthreads();

    unsigned lmin = 0x7F800000u, lmax = 0u;
#pragma unroll
    for (int r = 0; r < 4; ++r) {
        const int y = threadIdx.y + r * 8;
        const int x = threadIdx.x;
        const float c0 = tile[y + 4][x + 4];
        int mind2 = 1 << 30;
#pragma unroll
        for (int dy = -4; dy <= 4; ++dy) {
#pragma unroll
            for (int dx = -4; dx <= 4; ++dx) {
                if (dy == 0 && dx == 0) continue;
                if (tile[y + 4 + dy][x + 4 + dx] != c0) {
                    int d2 = dx * dx + dy * dy;
                    mind2 = d2 < mind2 ? d2 : mind2;
                }
            }
        }
        float contour = 0.0f;
        if (mind2 < (1 << 30)) contour = 1.0f / (sqrtf((float)mind2) + EPSF);
        const size_t gidx = (size_t)bz * HWn + (by + y) * Wn + (bx + x);
        float v = fwb[(by + y) * Wn + (bx + x)] + contour;
        v = v * v;
        wraw[gidx] = v;
        unsigned u = __float_as_uint(v);       // valid ordering: v >= 0
        lmin = u < lmin ? u : lmin;
        lmax = u > lmax ? u : lmax;
    }
    atomicMin(&smin, lmin);
    atomicMax(&smax, lmax);
    __syncthreads();
    if (tid == 0) { atomicMin(&minmax[0], smin); atomicMax(&minmax[1], smax); }
}

// ---------------- weighted softplus loss + WMMA wave reduction ------------
__global__ void __launch_bounds__(256) k_loss(const float* __restrict__ pred,
                                              const int* __restrict__ tgt,
                                              const float* __restrict__ wraw,
                                              const unsigned* __restrict__ minmax,
                                              float* __restrict__ partials) {
    const float wmin = __uint_as_float(minmax[0]);
    const float wmax = __uint_as_float(minmax[1]);
    const float scale = 2.0f / (wmax - wmin + EPSF);
    const int   tid0  = blockIdx.x * 256 + threadIdx.x;
    const int   stride = NBLK3 * 256;

    float a0 = 0.0f, a1 = 0.0f;                  // two f32 chains per lane
#pragma unroll 4
    for (int k = 0; k < ITER3; ++k) {            // trip count uniform: EXEC==all-1s
        int idx = tid0 + k * stride;
        int b     = idx / CHWn;
        int inner = idx - b * CHWn;
        int hw    = inner % HWn;
        float w = (wraw[(size_t)b * HWn + hw] - wmin) * scale;
        float x = pred[idx];
        float t = (float)tgt[idx];
        float sp = fmaxf(x, 0.0f) + log1pf(expf(-fabsf(x)));  // logaddexp(x,0)
        float v = (sp - t * x) * w;
        if (k & 1) a1 += v; else a0 += v;
    }

    float wavesum;
#if __has_builtin(__builtin_amdgcn_wmma_f32_16x16x4_f32)
    // A (16x4 f32) = all 64 lane-chain values of the wave; B = ones =>
    // D[m][n] = rowsum_m. Sum of each lane's 8 D regs covers half the rows;
    // one xor-16 shuffle completes the wave total. Layout-independent since
    // B is all ones.
    {
        v2f A = {a0, a1};
        v2f Bm = {1.0f, 1.0f};
        v8f Cz = {};
        v8f D = __builtin_amdgcn_wmma_f32_16x16x4_f32(
            /*neg_a=*/false, A, /*neg_b=*/false, Bm,
            /*c_mod=*/(short)0, Cz, /*reuse_a=*/false, /*reuse_b=*/false);
        float sown = 0.0f;
#pragma unroll
        for (int i = 0; i < 8; ++i) sown += D[i];
        wavesum = sown + __shfl_xor(sown, 16, 32);   // valid on all lanes
    }
#else
    {
        float v = a0 + a1;
#pragma unroll
        for (int off = 16; off > 0; off >>= 1) v += __shfl_down(v, off, 32);
        wavesum = v;                                  // valid on lane 0 only
    }
#endif

    __shared__ float wsum[8];                         // 8 waves per block
    const int lane = threadIdx.x & 31;
    const int wid  = threadIdx.x >> 5;
    if (lane == 0) wsum[wid] = wavesum;
    __syncthreads();
    if (threadIdx.x == 0) {
        float s = 0.0f;
#pragma unroll
        for (int i = 0; i < 8; ++i) s += wsum[i];
        partials[blockIdx.x] = s;                     // fixed assignment: deterministic
    }
}

// ---------------- final fixed-order reduction -----------------------------
__global__ void __launch_bounds__(256) k_final(const float* __restrict__ partials,
                                               float* __restrict__ out) {
    __shared__ float sm[256];
    float s = 0.0f;
    for (int i = threadIdx.x; i < NBLK3; i += 256) s += partials[i];
    sm[threadIdx.x] = s;
    __syncthreads();
    for (int off = 128; off > 0; off >>= 1) {
        if (threadIdx.x < off) sm[threadIdx.x] += sm[threadIdx.x + off];
        __syncthreads();
    }
    if (threadIdx.x == 0) out[0] = sm[0] * (1.0f / (float)BCHW);
}

extern "C" void kernel_launch(void* const* d_in, const int* in_sizes, int n_in,
                              void* d_out, int out_size, void* d_ws, size_t ws_size,
                              hipStream_t stream) {
    const float* pred = (const float*)d_in[0];
    const int*   tgt  = (const int*)d_in[1];

    // workspace layout (floats): fw[BHW] | wraw[BHW] | partials[NBLK3] | minmax[2]
    float*    fw       = (float*)d_ws;
    float*    wraw     = fw + BHW;
    float*    partials = wraw + BHW;
    unsigned* minmax   = (unsigned*)(partials + NBLK3);

    k_init   <<<1, 32, 0, stream>>>(minmax);
    k_channel<<<BHW / 256, 256, 0, stream>>>(tgt, fw);
    k_stencil<<<dim3(Wn / 32, Hn / 32, Bn), dim3(32, 8), 0, stream>>>(fw, wraw, minmax);
    k_loss   <<<NBLK3, 256, 0, stream>>>(pred, tgt, wraw, minmax, partials);
    k_final  <<<1, 256, 0, stream>>>(partials, (float*)d_out);
}